// RSE_63677185130930
// MI455X (gfx1250) — compile-verified
//
#include <hip/hip_runtime.h>
#include <hip/hip_bf16.h>
#include <math.h>

// ---------------- problem constants ----------------
#define B_SZV    2
#define SEQV     4096
#define D_MODELV 768
#define D_STATEV 128
#define D_CONVV  4
#define HEADDIMV 64
#define D_INNERV 1536
#define NHEADSV  24
#define D_XBCV   1792           // D_INNER + 2*D_STATE
#define D_PROJV  3352           // 2*D_INNER + 2*D_STATE + NHEADS
#define CHUNKV   128
#define NCHUNKV  32             // SEQ / CHUNK
#define EPSV     1e-5f
#define MTOK     (B_SZV*SEQV)   // 8192

typedef __attribute__((ext_vector_type(2))) float v2f;
typedef __attribute__((ext_vector_type(8))) float v8f;

// exact-f32 matrix op:  D(16x16) = A(16x4) * B(4x16) + C
#define WMMA_F32(a,b,c) \
  __builtin_amdgcn_wmma_f32_16x16x4_f32(false,(a),false,(b),(short)0,(c),false,false)

__device__ __forceinline__ float sigmoidf_(float x){ return 1.f/(1.f+__expf(-x)); }
__device__ __forceinline__ float siluf_(float x){ return x*sigmoidf_(x); }
__device__ __forceinline__ float softplusf_(float x){ return (x>20.f)? x : log1pf(__expf(x)); }

// =====================================================================
// Generic f32-WMMA GEMM:  Out[M,N] = X[M,K] @ W[N,K]^T  (row-major)
// Block tile 64(M) x 128(N), 8 waves, each wave a 32x32 tile (2x2 WMMA).
// EPI=true fuses: Out = sigmoid(gate[0])*acc + resid  (resid same shape).
// Requires: M % 64 == 0, K % 16 == 0. N guarded.
// =====================================================================
template<bool EPI>
__global__ __launch_bounds__(256)
void gemm_xwt(const float* __restrict__ X, const float* __restrict__ W,
              float* __restrict__ Out, int M, int N, int K,
              const float* __restrict__ gate, const float* __restrict__ resid)
{
  __shared__ float Xs[64][17];   // [m][k]
  __shared__ float Ws[128][17];  // [n][k]
  const int tid  = threadIdx.x;
  const int lane = tid & 31, wave = tid >> 5;
  const int m0 = blockIdx.y * 64;
  const int n0 = blockIdx.x * 128;
  const int wm = wave & 1, wn = wave >> 1;   // 2 x 4 wave grid
  const int lrow = lane & 15;
  const int ksel = (lane >> 4) * 2;          // lanes 0-15 -> K0/K1, 16-31 -> K2/K3

  v8f acc[2][2];
  for(int i=0;i<2;i++) for(int j=0;j<2;j++) acc[i][j] = (v8f)0.f;

  for(int k0=0; k0<K; k0+=16){
    __syncthreads();
    for(int t=tid; t<64*16; t+=256){
      int r=t>>4, c=t&15;
      Xs[r][c] = X[(size_t)(m0+r)*K + k0 + c];
    }
    for(int t=tid; t<128*16; t+=256){
      int r=t>>4, c=t&15;
      int n = n0 + r;
      Ws[r][c] = (n < N) ? W[(size_t)n*K + k0 + c] : 0.f;
    }
    if (k0 + 16 < K){ // speculative prefetch of next K-tile
      __builtin_prefetch(&X[(size_t)(m0 + (tid&63))*K + k0 + 16], 0, 0);
      int np = n0 + (tid>>1);
      if (np < N) __builtin_prefetch(&W[(size_t)np*K + k0 + 16], 0, 0);
    }
    __syncthreads();
    for(int kk=0; kk<16; kk+=4){
      for(int i=0;i<2;i++){
        v2f a;
        a.x = Xs[wm*32 + i*16 + lrow][kk + ksel];
        a.y = Xs[wm*32 + i*16 + lrow][kk + ksel + 1];
        for(int j=0;j<2;j++){
          v2f b;
          b.x = Ws[wn*32 + j*16 + lrow][kk + ksel];
          b.y = Ws[wn*32 + j*16 + lrow][kk + ksel + 1];
          acc[i][j] = WMMA_F32(a, b, acc[i][j]);
        }
      }
    }
  }

  float sg = 0.f;
  if constexpr (EPI) sg = sigmoidf_(gate[0]);
  for(int i=0;i<2;i++) for(int j=0;j<2;j++){
    int col = n0 + wn*32 + j*16 + lrow;
    if (col < N){
      for(int v=0;v<8;v++){
        int row = m0 + wm*32 + i*16 + v + 8*(lane>>4);
        size_t idx = (size_t)row*N + col;
        float val = acc[i][j][v];
        if constexpr (EPI) val = sg*val + resid[idx];
        Out[idx] = val;
      }
    }
  }
}

// =====================================================================
// Depthwise causal conv (K=4) + bias + SiLU, scatter into x / B / C
// =====================================================================
__global__ __launch_bounds__(256)
void conv_kernel(const float* __restrict__ zxbcdt,
                 const float* __restrict__ conv_w, const float* __restrict__ conv_b,
                 float* __restrict__ xcv, float* __restrict__ Bv, float* __restrict__ Cv)
{
  size_t idx = (size_t)blockIdx.x*256 + threadIdx.x;
  const size_t total = (size_t)MTOK * D_XBCV;
  if (idx >= total) return;
  int j = (int)(idx % D_XBCV);
  size_t bl = idx / D_XBCV;
  int l = (int)(bl % SEQV);
  size_t b = bl / SEQV;
  float acc = conv_b[j];
  #pragma unroll
  for(int k=0;k<D_CONVV;k++){
    int ls = l - (D_CONVV-1) + k;
    float v = (ls >= 0) ? zxbcdt[((size_t)b*SEQV + ls)*D_PROJV + D_INNERV + j] : 0.f;
    acc += conv_w[j*D_CONVV + k] * v;
  }
  acc = siluf_(acc);
  size_t tok = b*SEQV + l;
  if (j < D_INNERV)                 xcv[tok*D_INNERV + j] = acc;
  else if (j < D_INNERV + D_STATEV) Bv[tok*D_STATEV + (j - D_INNERV)] = acc;
  else                              Cv[tok*D_STATEV + (j - D_INNERV - D_STATEV)] = acc;
}

// dt = softplus(zxbcdt[..., -NHEADS:] + dt_bias)
__global__ __launch_bounds__(256)
void dt_kernel(const float* __restrict__ zxbcdt, const float* __restrict__ dt_bias,
               float* __restrict__ dtv)
{
  size_t idx = (size_t)blockIdx.x*256 + threadIdx.x;
  const size_t total = (size_t)MTOK * NHEADSV;
  if (idx >= total) return;
  int h = (int)(idx % NHEADSV);
  size_t bl = idx / NHEADSV;
  dtv[idx] = softplusf_(zxbcdt[bl*D_PROJV + (D_PROJV - NHEADSV) + h] + dt_bias[h]);
}

// Acum[(b,c,h)][l] = inclusive cumsum over l of dt*A  (A = -exp(A_log))
__global__ __launch_bounds__(256)
void acum_kernel(const float* __restrict__ dtv, const float* __restrict__ A_log,
                 float* __restrict__ Acum)
{
  int idx = blockIdx.x*256 + threadIdx.x;
  const int total = B_SZV*NCHUNKV*NHEADSV;
  if (idx >= total) return;
  int h = idx % NHEADSV; int bc = idx / NHEADSV;
  int c = bc % NCHUNKV;  int b = bc / NCHUNKV;
  float A = -__expf(A_log[h]);
  float s = 0.f;
  size_t base = ((size_t)b*SEQV + (size_t)c*CHUNKV)*NHEADSV + h;
  float* out = Acum + (size_t)idx*CHUNKV;
  for(int l=0;l<CHUNKV;l++){
    s += dtv[base + (size_t)l*NHEADSV] * A;
    out[l] = s;
  }
}

// =====================================================================
// states[(b,c,h)][p][n] = sum_l  x[l,p]*dt_l*exp(Alast-Acum_l) * B[l,n]
// 64x128 output via WMMA; one block per (b,c,h).
// =====================================================================
__global__ __launch_bounds__(256)
void states_kernel(const float* __restrict__ xcv, const float* __restrict__ Bv,
                   const float* __restrict__ dtv, const float* __restrict__ Acum,
                   float* __restrict__ states)
{
  __shared__ float Xw[16][64];    // [k=l][p]  (already scaled by w_l)
  __shared__ float Bs[16][128];   // [k=l][n]
  __shared__ float wl[CHUNKV];
  int blk = blockIdx.x;           // (b*NCHUNK + c)*NHEADS + h
  int h = blk % NHEADSV; int bc = blk / NHEADSV;
  int c = bc % NCHUNKV;  int b = bc / NCHUNKV;
  int tid = threadIdx.x, lane = tid & 31, wave = tid >> 5;
  int lrow = lane & 15, ksel = (lane>>4)*2;
  size_t tokbase = (size_t)b*SEQV + (size_t)c*CHUNKV;
  const float* ac = Acum + (size_t)blk*CHUNKV;

  if (tid < CHUNKV){
    float alast = ac[CHUNKV-1];
    wl[tid] = __expf(alast - ac[tid]) * dtv[(tokbase + tid)*NHEADSV + h];
  }
  int wp = wave & 3, wn2 = wave >> 2;  // wave: 16 p-rows x 64 n-cols
  v8f acc[4]; for(int j=0;j<4;j++) acc[j] = (v8f)0.f;

  for(int k0=0;k0<CHUNKV;k0+=16){
    __syncthreads();
    for(int t=tid;t<16*64;t+=256){
      int k=t>>6, p=t&63; int l=k0+k;
      Xw[k][p] = xcv[((tokbase + l)*NHEADSV + h)*HEADDIMV + p] * wl[l];
    }
    for(int t=tid;t<16*128;t+=256){
      int k=t>>7, n=t&127;
      Bs[k][n] = Bv[(tokbase + k0 + k)*D_STATEV + n];
    }
    __syncthreads();
    for(int kk=0;kk<16;kk+=4){
      v2f a;
      a.x = Xw[kk+ksel  ][wp*16 + lrow];
      a.y = Xw[kk+ksel+1][wp*16 + lrow];
      for(int j=0;j<4;j++){
        v2f bb;
        bb.x = Bs[kk+ksel  ][wn2*64 + j*16 + lrow];
        bb.y = Bs[kk+ksel+1][wn2*64 + j*16 + lrow];
        acc[j] = WMMA_F32(a, bb, acc[j]);
      }
    }
  }
  float* st = states + (size_t)blk*HEADDIMV*D_STATEV;
  for(int j=0;j<4;j++){
    int n = wn2*64 + j*16 + lrow;
    for(int v=0;v<8;v++){
      int p = wp*16 + v + 8*(lane>>4);
      st[p*D_STATEV + n] = acc[j][v];
    }
  }
}

// Inter-chunk scan (in place): states[c] becomes prev-state S_c;
// S_{c+1} = exp(A_last_c)*S_c + states_c.  One block per (b,h).
__global__ __launch_bounds__(256)
void scan_kernel(float* __restrict__ states, const float* __restrict__ Acum)
{
  int bh = blockIdx.x;
  int h = bh % NHEADSV, b = bh / NHEADSV;
  int tid = threadIdx.x;
  const int PN = HEADDIMV*D_STATEV; // 8192
  float S[32];
  #pragma unroll
  for(int i=0;i<32;i++) S[i] = 0.f;
  for(int c=0;c<NCHUNKV;c++){
    size_t blk = (size_t)(b*NCHUNKV + c)*NHEADSV + h;
    float e = __expf(Acum[blk*CHUNKV + CHUNKV-1]);
    float* st = states + blk*PN;
    #pragma unroll
    for(int i=0;i<32;i++){
      int idx = tid + i*256;
      float cur = st[idx];
      st[idx] = S[i];
      S[i] = e*S[i] + cur;
    }
  }
}

// =====================================================================
// Y[(b,c)][l][h][p] = (L .* (C B^T)) @ (x*dt)  +  exp(Acum)*C @ prev^T + D*x
// One block per (b,c,h); CB (128x128) materialized in dynamic LDS.
// =====================================================================
#define CBSTRIDE 130
__global__ __launch_bounds__(256)
void ychunk_kernel(const float* __restrict__ xcv, const float* __restrict__ Bv,
                   const float* __restrict__ Cv,  const float* __restrict__ dtv,
                   const float* __restrict__ Acum, const float* __restrict__ states,
                   const float* __restrict__ Dv,  float* __restrict__ Y)
{
  extern __shared__ float sm[];
  float* CBs = sm;                       // 128*130
  float* Cs  = CBs + 128*CBSTRIDE;       // 128*17  (A-operand staging)
  float* Bst = Cs  + 128*17;             // 2048    (B-operand staging)
  float* aAc = Bst + 2048;               // 128
  float* adt = aAc + 128;                // 128
  float* aeA = adt + 128;                // 128

  int blk = blockIdx.x;
  int h = blk % NHEADSV; int bc = blk / NHEADSV;
  int c = bc % NCHUNKV;  int b = bc / NCHUNKV;
  int tid = threadIdx.x, lane = tid & 31, wave = tid >> 5;
  int lrow = lane & 15, ksel = (lane>>4)*2;
  size_t tokbase = (size_t)b*SEQV + (size_t)c*CHUNKV;

  if (tid < CHUNKV){
    float a = Acum[(size_t)blk*CHUNKV + tid];
    aAc[tid] = a;
    aeA[tid] = __expf(a);
    adt[tid] = dtv[(tokbase + tid)*NHEADSV + h];
  }
  __syncthreads();

  // ---- Phase 1: CB[l][s] = sum_n C[l,n]*B[s,n] -> LDS ----
  {
    v8f p1[8]; for(int j=0;j<8;j++) p1[j] = (v8f)0.f;
    for(int n0=0;n0<D_STATEV;n0+=16){
      __syncthreads();
      for(int t=tid;t<128*16;t+=256){
        int l=t>>4, k=t&15;
        Cs[l*17+k] = Cv[(tokbase + l)*D_STATEV + n0 + k];
      }
      for(int t=tid;t<16*128;t+=256){
        int k=t>>7, s=t&127;
        Bst[k*128+s] = Bv[(tokbase + s)*D_STATEV + n0 + k];
      }
      __syncthreads();
      for(int kk=0;kk<16;kk+=4){
        v2f a;
        a.x = Cs[(wave*16+lrow)*17 + kk+ksel];
        a.y = Cs[(wave*16+lrow)*17 + kk+ksel+1];
        for(int j=0;j<8;j++){
          v2f bb;
          bb.x = Bst[(kk+ksel  )*128 + j*16+lrow];
          bb.y = Bst[(kk+ksel+1)*128 + j*16+lrow];
          p1[j] = WMMA_F32(a, bb, p1[j]);
        }
      }
    }
    __syncthreads();
    for(int j=0;j<8;j++){
      int s = j*16 + lrow;
      for(int v=0;v<8;v++){
        int l = wave*16 + v + 8*(lane>>4);
        CBs[l*CBSTRIDE + s] = p1[j][v];
      }
    }
    __syncthreads();
  }

  // ---- Phase 2: Y += (mask.*CB.*exp(dAcum).*dt_s) @ x   (K = s) ----
  v8f acc[4]; for(int j=0;j<4;j++) acc[j] = (v8f)0.f;
  for(int s0=0;s0<CHUNKV;s0+=16){
    __syncthreads();
    for(int t=tid;t<16*64;t+=256){
      int k=t>>6, p=t&63;
      Bst[k*64+p] = xcv[((tokbase + s0 + k)*NHEADSV + h)*HEADDIMV + p];
    }
    __syncthreads();
    for(int kk=0;kk<16;kk+=4){
      int l = wave*16 + lrow;
      int s1 = s0 + kk + ksel, s2 = s1 + 1;
      float al = aAc[l];
      v2f a;
      a.x = (s1<=l) ? CBs[l*CBSTRIDE+s1]*__expf(al - aAc[s1])*adt[s1] : 0.f;
      a.y = (s2<=l) ? CBs[l*CBSTRIDE+s2]*__expf(al - aAc[s2])*adt[s2] : 0.f;
      for(int j=0;j<4;j++){
        v2f bb;
        bb.x = Bst[(kk+ksel  )*64 + j*16+lrow];
        bb.y = Bst[(kk+ksel+1)*64 + j*16+lrow];
        acc[j] = WMMA_F32(a, bb, acc[j]);
      }
    }
  }

  // ---- Phase 3: Y += (exp(Acum_l)*C) @ prev^T   (K = n) ----
  const float* prev = states + (size_t)blk*HEADDIMV*D_STATEV;
  for(int n0=0;n0<D_STATEV;n0+=16){
    __syncthreads();
    for(int t=tid;t<128*16;t+=256){
      int l=t>>4, k=t&15;
      Cs[l*17+k] = Cv[(tokbase + l)*D_STATEV + n0 + k];
    }
    for(int t=tid;t<16*64;t+=256){
      int k=t>>6, p=t&63;
      Bst[k*64+p] = prev[p*D_STATEV + n0 + k];
    }
    __syncthreads();
    for(int kk=0;kk<16;kk+=4){
      int l = wave*16 + lrow;
      float e = aeA[l];
      v2f a;
      a.x = Cs[l*17 + kk+ksel  ]*e;
      a.y = Cs[l*17 + kk+ksel+1]*e;
      for(int j=0;j<4;j++){
        v2f bb;
        bb.x = Bst[(kk+ksel  )*64 + j*16+lrow];
        bb.y = Bst[(kk+ksel+1)*64 + j*16+lrow];
        acc[j] = WMMA_F32(a, bb, acc[j]);
      }
    }
  }

  // ---- Epilogue: + D[h]*x, store Y[b,l,h,p] ----
  float Dh = Dv[h];
  for(int j=0;j<4;j++){
    int p = j*16 + lrow;
    for(int v=0;v<8;v++){
      int l = wave*16 + v + 8*(lane>>4);
      size_t xi = ((tokbase + l)*NHEADSV + h)*HEADDIMV + p;
      Y[xi] = acc[j][v] + Dh*xcv[xi];
    }
  }
}

// =====================================================================
// y = Y * silu(z);  RMSNorm;  one block per token
// =====================================================================
__global__ __launch_bounds__(256)
void norm_kernel(const float* __restrict__ Y, const float* __restrict__ zxbcdt,
                 const float* __restrict__ norm_w, float* __restrict__ ynorm)
{
  __shared__ float red[8];
  size_t tok = blockIdx.x;
  int tid = threadIdx.x;
  const float* y = Y + tok*D_INNERV;
  const float* z = zxbcdt + tok*D_PROJV;
  float g[6];
  float ss = 0.f;
  #pragma unroll
  for(int i=0;i<6;i++){
    int j = tid + i*256;
    float v = y[j] * siluf_(z[j]);
    g[i] = v; ss += v*v;
  }
  for(int off=16; off>0; off>>=1) ss += __shfl_down(ss, off, 32);
  if ((tid & 31) == 0) red[tid>>5] = ss;
  __syncthreads();
  if (tid < 8){
    float v = red[tid];
    for(int off=4; off>0; off>>=1) v += __shfl_down(v, off, 8);
    if (tid == 0) red[0] = v;
  }
  __syncthreads();
  float inv = rsqrtf(red[0]/(float)D_INNERV + EPSV);
  #pragma unroll
  for(int i=0;i<6;i++){
    int j = tid + i*256;
    ynorm[tok*D_INNERV + j] = g[i]*inv*norm_w[j];
  }
}

// =====================================================================
extern "C" void kernel_launch(void* const* d_in, const int* in_sizes, int n_in,
                              void* d_out, int out_size, void* d_ws, size_t ws_size,
                              hipStream_t stream)
{
  const float* feature    = (const float*)d_in[0];
  const float* gate1      = (const float*)d_in[1];
  const float* in_proj_w  = (const float*)d_in[2];
  const float* conv_w     = (const float*)d_in[3];
  const float* conv_b     = (const float*)d_in[4];
  const float* dt_bias    = (const float*)d_in[5];
  const float* A_log      = (const float*)d_in[6];
  const float* Dv         = (const float*)d_in[7];
  const float* norm_w     = (const float*)d_in[8];
  const float* out_proj_w = (const float*)d_in[9];
  float* out = (float*)d_out;
  float* ws  = (float*)d_ws;

  // workspace layout (floats)
  size_t o = 0;
  float* zx    = ws + o; o += (size_t)MTOK * D_PROJV;            // 27.46M
  float* xcv   = ws + o; o += (size_t)MTOK * D_INNERV;           // 12.58M
  float* Bv    = ws + o; o += (size_t)MTOK * D_STATEV;           // 1.05M
  float* Cv    = ws + o; o += (size_t)MTOK * D_STATEV;           // 1.05M
  float* dtv   = ws + o; o += (size_t)MTOK * NHEADSV;            // 0.20M
  float* Acum  = ws + o; o += (size_t)B_SZV*NCHUNKV*NHEADSV*CHUNKV; // 0.20M
  float* stat  = ws + o; o += (size_t)B_SZV*NCHUNKV*NHEADSV*HEADDIMV*D_STATEV; // 12.58M
  float* Yb    = ws + o; o += (size_t)MTOK * D_INNERV;           // 12.58M
  float* ynorm = xcv;  // x no longer needed after ychunk_kernel

  // 1) in_proj GEMM: [8192,3352] = [8192,768] @ [3352,768]^T
  gemm_xwt<false><<<dim3((D_PROJV+127)/128, MTOK/64), 256, 0, stream>>>(
      feature, in_proj_w, zx, MTOK, D_PROJV, D_MODELV, nullptr, nullptr);

  // 2) conv + silu
  {
    size_t total = (size_t)MTOK * D_XBCV;
    conv_kernel<<<(unsigned)((total+255)/256), 256, 0, stream>>>(zx, conv_w, conv_b, xcv, Bv, Cv);
  }
  // 3) dt
  {
    size_t total = (size_t)MTOK * NHEADSV;
    dt_kernel<<<(unsigned)((total+255)/256), 256, 0, stream>>>(zx, dt_bias, dtv);
  }
  // 4) per-chunk cumsum of dt*A
  acum_kernel<<<(B_SZV*NCHUNKV*NHEADSV + 255)/256, 256, 0, stream>>>(dtv, A_log, Acum);

  // 5) chunk states (WMMA)
  states_kernel<<<B_SZV*NCHUNKV*NHEADSV, 256, 0, stream>>>(xcv, Bv, dtv, Acum, stat);

  // 6) inter-chunk scan (states -> prev, in place)
  scan_kernel<<<B_SZV*NHEADSV, 256, 0, stream>>>(stat, Acum);

  // 7) Y chunks (WMMA x3 phases), dynamic LDS
  {
    size_t smem = (size_t)(128*CBSTRIDE + 128*17 + 2048 + 3*128) * sizeof(float); // ~85 KB
    ychunk_kernel<<<B_SZV*NCHUNKV*NHEADSV, 256, smem, stream>>>(
        xcv, Bv, Cv, dtv, Acum, stat, Dv, Yb);
  }

  // 8) gating + RMSNorm
  norm_kernel<<<MTOK, 256, 0, stream>>>(Yb, zx, norm_w, ynorm);

  // 9) out_proj GEMM + fused sigmoid(gate1)*out + feature
  gemm_xwt<true><<<dim3((D_MODELV+127)/128, MTOK/64), 256, 0, stream>>>(
      ynorm, out_proj_w, out, MTOK, D_MODELV, D_INNERV, gate1, feature);
}